// CELAB_84344567758977
// MI455X (gfx1250) — compile-verified
//
#include <hip/hip_runtime.h>

// ============================================================================
// CELAB on MI455X (gfx1250, wave32).  All matmuls use V_WMMA_F32_16X16X4_F32.
//
// Input ordering assumption (setup_inputs() dict insertion order, params
// nested dicts in insertion order):
//   0:x 1:y
//   2..5   lfe_x0 {w0,b0,w1,b1}   6..9  lfe_x1   10..13 lfe_y0   14..17 lfe_y1
//   18..25 gmsa0  {wi,bi,g,bt,m,v,wo,bo}
//   26..33 gmsa1  34..41 cgmsa0   42..49 cgmsa1
//
// Workspace: ~370 MB (4x 60ch tensors, 2x 120ch proj buffers, ys buffer,
// 3x compact q-save buffers, 1 compact v buffer).
// ============================================================================

typedef __attribute__((ext_vector_type(2))) float v2f;
typedef __attribute__((ext_vector_type(8))) float v8f;

#define HW    256
#define BATCH 2
#define NPIX  (BATCH * HW * HW)   // 131072 pixels (= tokens)
#define DH    20                  // per-window q/v channel count

// token/window -> planar NCHW index, with the +ws/2 roll folded in (mod 256)
__device__ __forceinline__ size_t tok_plane(int win, int t, int ch, int C, int ws) {
  int nb = HW / ws;
  int b  = win / (nb * nb);
  int r  = win - b * nb * nb;
  int hb = r / nb;
  int wb = r - hb * nb;
  int ti = t / ws;
  int tj = t - ti * ws;
  int ii = (hb * ws + ti + (ws >> 1)) & (HW - 1);
  int jj = (wb * ws + tj + (ws >> 1)) & (HW - 1);
  return (((size_t)b * C + ch) * HW + ii) * HW + jj;
}

// ============================================================================
// Fused 1x1 conv: out = [relu]( BN( W @ shift5(in) + bias ) ) [+ res]
// GEMM: M=cout (<=120, padded to 16 in LDS), K=CIN (60/120, exact mult of 4),
// N=16 pixels/wave, 4 pixel tiles per wave.  All WMMA-feeding loads are
// unconditional (padded LDS rows, wrapped+selected border loads) so the
// compiler emits plain ds_load_b64 / global_load with no exec masking.
// ============================================================================
template <int CIN>
__global__ __launch_bounds__(128) void conv1x1_kernel(
    const float* __restrict__ in, float* __restrict__ out,
    const float* __restrict__ Wt, const float* __restrict__ bias,
    const float* __restrict__ bn_g, const float* __restrict__ bn_b,
    const float* __restrict__ bn_m, const float* __restrict__ bn_v,
    const float* __restrict__ res, int cout, int do_shift, int do_relu)
{
  __shared__ float sW[128 * CIN];   // M padded to multiple of 16, zero-filled
  __shared__ float sB[128];
  const int tid = threadIdx.x;

  // stage BN-folded weights + bias into LDS (padded rows = 0)
  const int mtiles = (cout + 15) >> 4;
  const int padm   = mtiles << 4;
  const int nw     = padm * CIN;
  for (int idx = tid; idx < nw; idx += 128) {
    int o = idx / CIN;
    float w = 0.f;
    if (o < cout) {
      float s = bn_g ? bn_g[o] * rsqrtf(bn_v[o] + 1e-5f) : 1.f;
      w = Wt[idx] * s;
    }
    sW[idx] = w;
  }
  for (int o = tid; o < cout; o += 128) {
    float b = bias[o];
    if (bn_g) {
      float s = bn_g[o] * rsqrtf(bn_v[o] + 1e-5f);
      b = (b - bn_m[o]) * s + bn_b[o];
    }
    sB[o] = b;
  }
  __syncthreads();

  const int wave = tid >> 5;
  const int lane = tid & 31;
  const int n    = lane & 15;   // pixel within tile / output column
  const int kh   = lane >> 4;   // K-pair selector

  constexpr int NF  = CIN / 4;        // K-steps (exact: 15 or 30)
  constexpr int GSZ = CIN / 5;        // shift5 group size (12 or 24)

  for (int tile = 0; tile < 4; ++tile) {
    const int p0  = (((blockIdx.x * 4 + wave) * 4) + tile) * 16;
    const int b   = p0 >> 16;           // 65536 pixels per batch image
    const int rem = p0 & 65535;
    const int i   = rem >> 8;
    const int j   = (rem & 255) + n;

    // load activation K-panel (shift5 folded in; branchless border zeroing)
    v2f bf[NF];
#pragma unroll
    for (int f = 0; f < NF; ++f) {
#pragma unroll
      for (int e = 0; e < 2; ++e) {
        int k  = f * 4 + kh * 2 + e;
        int di = 0, dj = 0;
        if (do_shift) {
          int grp = k / GSZ;            // 0..4
          if      (grp == 0) dj =  1;
          else if (grp == 1) dj = -1;
          else if (grp == 2) di =  1;
          else if (grp == 3) di = -1;
        }
        int  ii  = i + di, jj = j + dj;
        bool inb = ((unsigned)ii < (unsigned)HW) & ((unsigned)jj < (unsigned)HW);
        float raw =
            in[(((size_t)b * CIN + k) * HW + (ii & (HW - 1))) * HW + (jj & (HW - 1))];
        bf[f][e] = inb ? raw : 0.f;
      }
    }

    for (int mt = 0; mt < mtiles; ++mt) {
      v8f acc = {};
      const int mm = mt * 16 + (lane & 15);    // < padm, always valid in sW
#pragma unroll
      for (int f = 0; f < NF; ++f) {
        int k = f * 4 + kh * 2;
        v2f af;
        af[0] = sW[mm * CIN + k];
        af[1] = sW[mm * CIN + k + 1];
        acc = __builtin_amdgcn_wmma_f32_16x16x4_f32(
            false, af, false, bf[f], (short)0, acc, false, false);
      }
#pragma unroll
      for (int r = 0; r < 8; ++r) {
        int o = mt * 16 + r + 8 * kh;
        if (o < cout) {
          float val = acc[r] + sB[o];
          if (do_relu) val = fmaxf(val, 0.f);
          size_t oi = (((size_t)b * cout + o) * HW + i) * HW + j;
          if (res) val += res[oi];
          out[oi] = val;
        }
      }
    }
  }
}

// ============================================================================
// Gather 20 channels of a planar buffer into compact windowed layout:
//   dst[(win*L + t)*20 + c]   (roll folded in) -- makes WMMA fragment loads
// contiguous, and doubles as the persistent q-save used by the apply stages.
// ============================================================================
__global__ __launch_bounds__(256) void win_gather_kernel(
    const float* __restrict__ src, float* __restrict__ dst,
    int srcC, int ch_off, int ws)
{
  int idx = blockIdx.x * blockDim.x + threadIdx.x;
  if (idx >= NPIX * DH) return;
  int c   = idx % DH;
  int tok = idx / DH;
  int L   = ws * ws;
  int t   = tok % L;
  int win = tok / L;
  dst[idx] = src[tok_plane(win, t, ch_off + c, srcC, ws)];
}

// ============================================================================
// Windowed attention: out = softmax(qx @ qy^T) @ v, scattered into the planar
// ys buffer (roll-back folded into the scatter).  One wave per 16-row S-tile;
// total row-tiles = NPIX/16 = 8192 for every ws.  S tile lives in LDS.
// ============================================================================
template <int WS, int WAVES>
__global__ __launch_bounds__(WAVES * 32) void attn_kernel(
    const float* __restrict__ qx, const float* __restrict__ qy,
    const float* __restrict__ vv, float* __restrict__ ys, int ys_coff)
{
  constexpr int L   = WS * WS;
  constexpr int TPW = L / 16;     // row-tiles per window
  extern __shared__ float smem[];

  const int tid  = threadIdx.x;
  const int wave = tid >> 5;
  const int lane = tid & 31;
  const int n    = lane & 15;
  const int kh   = lane >> 4;

  const int rt  = blockIdx.x * WAVES + wave;
  const int win = rt / TPW;
  const int tb  = (rt - win * TPW) * 16;    // row-token base
  float* S = smem + wave * (16 * L);
  const size_t qbase = (size_t)win * L;

  // A fragments: this tile's 16 q rows (K = DH = 20 -> 5 steps)
  v2f qa[5];
#pragma unroll
  for (int f = 0; f < 5; ++f) {
    const float* p = qx + (qbase + tb + n) * DH + f * 4 + kh * 2;
    qa[f][0] = p[0]; qa[f][1] = p[1];
  }

  // S = qx @ qy^T  (16 x L)
  for (int nt = 0; nt < TPW; ++nt) {
    v8f acc = {};
#pragma unroll
    for (int f = 0; f < 5; ++f) {
      const float* p = qy + (qbase + nt * 16 + n) * DH + f * 4 + kh * 2;
      v2f bfr; bfr[0] = p[0]; bfr[1] = p[1];
      acc = __builtin_amdgcn_wmma_f32_16x16x4_f32(
          false, qa[f], false, bfr, (short)0, acc, false, false);
    }
#pragma unroll
    for (int r = 0; r < 8; ++r)
      S[(r + 8 * kh) * L + nt * 16 + n] = acc[r];
  }
  __syncthreads();

  // row-wise softmax over L columns (wave32 shuffle reductions)
  for (int r = 0; r < 16; ++r) {
    float* row = S + r * L;
    float mx = -3.4e38f;
    for (int idx = lane; idx < L; idx += 32) mx = fmaxf(mx, row[idx]);
#pragma unroll
    for (int off = 16; off; off >>= 1) mx = fmaxf(mx, __shfl_xor(mx, off, 32));
    float sum = 0.f;
    for (int idx = lane; idx < L; idx += 32) {
      float e = __expf(row[idx] - mx);
      row[idx] = e;
      sum += e;
    }
#pragma unroll
    for (int off = 16; off; off >>= 1) sum += __shfl_xor(sum, off, 32);
    float inv = 1.f / sum;
    for (int idx = lane; idx < L; idx += 32) row[idx] *= inv;
  }
  __syncthreads();

  // out = S @ v   (M=16 rows, N=20 channels -> 2 n-tiles, K=L)
  // v loads use a clamped channel so they are unconditional; the garbage
  // columns (ch >= 20) are computed but never stored.
#pragma unroll
  for (int nt2 = 0; nt2 < 2; ++nt2) {
    const int ch  = nt2 * 16 + n;
    const int chc = ch < DH ? ch : DH - 1;
    v8f acc = {};
    for (int kf = 0; kf < L / 4; ++kf) {
      int k = kf * 4 + kh * 2;
      v2f afr; afr[0] = S[n * L + k]; afr[1] = S[n * L + k + 1];
      v2f bfr;
      bfr[0] = vv[(qbase + k)     * DH + chc];
      bfr[1] = vv[(qbase + k + 1) * DH + chc];
      acc = __builtin_amdgcn_wmma_f32_16x16x4_f32(
          false, afr, false, bfr, (short)0, acc, false, false);
    }
    if (ch < DH) {
#pragma unroll
      for (int r = 0; r < 8; ++r) {
        int t = tb + r + 8 * kh;
        ys[tok_plane(win, t, ys_coff + ch, 60, WS)] = acc[r];
      }
    }
  }
}

// ============================================================================
// Host-side orchestration
// ============================================================================
static inline void conv_launch(hipStream_t st, const float* in, float* out,
                               const float* W, const float* b,
                               const float* g, const float* bt,
                               const float* m, const float* v,
                               const float* res, int cin, int cout,
                               int shift, int relu) {
  if (cin == 60)
    conv1x1_kernel<60><<<NPIX / 256, 128, 0, st>>>(in, out, W, b, g, bt, m, v,
                                                   res, cout, shift, relu);
  else
    conv1x1_kernel<120><<<NPIX / 256, 128, 0, st>>>(in, out, W, b, g, bt, m, v,
                                                    res, cout, shift, relu);
}

static inline void gather_launch(hipStream_t st, const float* src, float* dst,
                                 int srcC, int coff, int ws) {
  int total = NPIX * DH;
  win_gather_kernel<<<(total + 255) / 256, 256, 0, st>>>(src, dst, srcC, coff, ws);
}

static inline void attn_launch(hipStream_t st, int ws, const float* qx,
                               const float* qy, const float* v, float* ys,
                               int coff) {
  if (ws == 4)
    attn_kernel<4, 4><<<2048, 128, 4 * 16 * 16 * sizeof(float), st>>>(qx, qy, v, ys, coff);
  else if (ws == 8)
    attn_kernel<8, 4><<<2048, 128, 4 * 16 * 64 * sizeof(float), st>>>(qx, qy, v, ys, coff);
  else
    attn_kernel<16, 2><<<4096, 64, 2 * 16 * 256 * sizeof(float), st>>>(qx, qy, v, ys, coff);
}

extern "C" void kernel_launch(void* const* d_in, const int* in_sizes, int n_in,
                              void* d_out, int out_size, void* d_ws, size_t ws_size,
                              hipStream_t stream) {
  (void)in_sizes; (void)n_in; (void)out_size; (void)ws_size;
  const float* X = (const float*)d_in[0];
  const float* Y = (const float*)d_in[1];
#define PF(i) ((const float*)d_in[i])

  const size_t T  = (size_t)BATCH * 60 * HW * HW;    // 60-ch tensor
  const size_t TP = (size_t)BATCH * 120 * HW * HW;   // 120-ch tensor
  const size_t TQ = (size_t)NPIX * DH;               // compact windowed 20-ch

  float* wsf = (float*)d_ws;
  float* xA  = wsf;        float* xB  = xA + T;
  float* yA  = xB + T;     float* yB  = yA + T;
  float* PB1 = yB + T;     float* PB2 = PB1 + TP;
  float* YS  = PB2 + TP;
  float* QGX = YS + T;                 // gmsa0 saved q       (3*TQ)
  float* QCX = QGX + 3 * TQ;           // cgmsa0 saved qx     (3*TQ)
  float* QCY = QCX + 3 * TQ;           // cgmsa0 saved qy     (3*TQ)
  float* VT  = QCY + 3 * TQ;           // transient compact v (TQ)
  float* OX  = (float*)d_out;
  float* OY  = OX + T;
  const int WSZ[3] = {4, 8, 16};

  // 1) x1 = lfe_x0(x) + x
  conv_launch(stream, X,   PB1, PF(2), PF(3), nullptr,nullptr,nullptr,nullptr, nullptr, 60, 120, 1, 1);
  conv_launch(stream, PB1, xA,  PF(4), PF(5), nullptr,nullptr,nullptr,nullptr, X,      120,  60, 1, 0);

  // 2) gmsa0 (calc): x2 = wo(attn(proj(x1))) + x1 ; save q windows
  conv_launch(stream, xA, PB1, PF(18), PF(19), PF(20), PF(21), PF(22), PF(23), nullptr, 60, 120, 0, 0);
  for (int i = 0; i < 3; ++i) {
    int ws = WSZ[i];
    gather_launch(stream, PB1, QGX + i * TQ, 120, i * 40,      ws);
    gather_launch(stream, PB1, VT,           120, i * 40 + 20, ws);
    attn_launch(stream, ws, QGX + i * TQ, QGX + i * TQ, VT, YS, i * 20);
  }
  conv_launch(stream, YS, xB, PF(24), PF(25), nullptr,nullptr,nullptr,nullptr, xA, 60, 60, 0, 0);

  // 3) y1 = lfe_y0(y) + y
  conv_launch(stream, Y,   PB1, PF(10), PF(11), nullptr,nullptr,nullptr,nullptr, nullptr, 60, 120, 1, 1);
  conv_launch(stream, PB1, yA,  PF(12), PF(13), nullptr,nullptr,nullptr,nullptr, Y,      120,  60, 1, 0);

  // 4) cgmsa0 (calc): y2 = wo(cross_attn(proj(x2), proj(y1))) + y1 ; save qx,qy
  conv_launch(stream, xB, PB1, PF(34), PF(35), PF(36), PF(37), PF(38), PF(39), nullptr, 60, 120, 0, 0);
  conv_launch(stream, yA, PB2, PF(34), PF(35), PF(36), PF(37), PF(38), PF(39), nullptr, 60, 120, 0, 0);
  for (int i = 0; i < 3; ++i) {
    int ws = WSZ[i];
    gather_launch(stream, PB1, QCX + i * TQ, 120, i * 40,      ws);
    gather_launch(stream, PB2, QCY + i * TQ, 120, i * 40,      ws);
    gather_launch(stream, PB1, VT,           120, i * 40 + 20, ws);
    attn_launch(stream, ws, QCX + i * TQ, QCY + i * TQ, VT, YS, i * 20);
  }
  conv_launch(stream, YS, yB, PF(40), PF(41), nullptr,nullptr,nullptr,nullptr, yA, 60, 60, 0, 0);

  // 5) x3 = lfe_x1(x2) + x2
  conv_launch(stream, xB,  PB1, PF(6), PF(7), nullptr,nullptr,nullptr,nullptr, nullptr, 60, 120, 1, 1);
  conv_launch(stream, PB1, xA,  PF(8), PF(9), nullptr,nullptr,nullptr,nullptr, xB,     120,  60, 1, 0);

  // 6) x4 = gmsa_apply(x3, atn_x, gmsa1) + x3  -> OX  (recompute softmax from QGX)
  conv_launch(stream, xA, PB1, PF(26), PF(27), PF(28), PF(29), PF(30), PF(31), nullptr, 60, 60, 0, 0);
  for (int i = 0; i < 3; ++i) {
    int ws = WSZ[i];
    gather_launch(stream, PB1, VT, 60, i * 20, ws);
    attn_launch(stream, ws, QGX + i * TQ, QGX + i * TQ, VT, YS, i * 20);
  }
  conv_launch(stream, YS, OX, PF(32), PF(33), nullptr,nullptr,nullptr,nullptr, xA, 60, 60, 0, 0);

  // 7) y3 = lfe_y1(y2) + y2
  conv_launch(stream, yB,  PB1, PF(14), PF(15), nullptr,nullptr,nullptr,nullptr, nullptr, 60, 120, 1, 1);
  conv_launch(stream, PB1, yA,  PF(16), PF(17), nullptr,nullptr,nullptr,nullptr, yB,     120,  60, 1, 0);

  // 8) y4 = cgmsa_apply(x4, atn_c, cgmsa1) + y3 -> OY (recompute from QCX/QCY)
  conv_launch(stream, OX, PB1, PF(42), PF(43), PF(44), PF(45), PF(46), PF(47), nullptr, 60, 60, 0, 0);
  for (int i = 0; i < 3; ++i) {
    int ws = WSZ[i];
    gather_launch(stream, PB1, VT, 60, i * 20, ws);
    attn_launch(stream, ws, QCX + i * TQ, QCY + i * TQ, VT, YS, i * 20);
  }
  conv_launch(stream, YS, OY, PF(48), PF(49), nullptr,nullptr,nullptr,nullptr, yA, 60, 60, 0, 0);
#undef PF
}